// OctMLP_55886114455795
// MI455X (gfx1250) — compile-verified
//
#include <hip/hip_runtime.h>

typedef float v2f __attribute__((ext_vector_type(2)));
typedef float v4f __attribute__((ext_vector_type(4)));
typedef float v8f __attribute__((ext_vector_type(8)));

#define LAMBDA_ 0.01f
#define B_ 8
#define S_ 1024
#define E_ 512
#define BS_ (B_ * S_)

// ---------------------------------------------------------------------------
// Kernel 1: row-sum of each 512x512 f32 weight matrix via V_WMMA_F32_16X16X4_F32
// (Y @ ones). One wave (32 threads) per 16-row block; 8 matrices * 32 blocks.
// A-layout (ISA 7.12.2, 32-bit A 16x4): lanes 0-15 -> K={0,1}, lanes 16-31 ->
// K={2,3}; M = lane & 15. B = all ones (layout irrelevant). Every column of
// the 16x16 D tile holds the accumulated row sums; lane 0 carries M=0..7 in
// VGPRs 0..7 and lane 16 carries M=8..15.
// ---------------------------------------------------------------------------
__global__ void __launch_bounds__(32) rowsum_wmma(
    const float* __restrict__ y0, const float* __restrict__ y1,
    const float* __restrict__ y2, const float* __restrict__ y3,
    const float* __restrict__ y4, const float* __restrict__ y5,
    const float* __restrict__ y6, const float* __restrict__ y7,
    float* __restrict__ ws)
{
    const int m  = blockIdx.x >> 5;   // which matrix
    const int rb = blockIdx.x & 31;   // which 16-row block
    const float* Y =
        (m == 0) ? y0 : (m == 1) ? y1 : (m == 2) ? y2 : (m == 3) ? y3 :
        (m == 4) ? y4 : (m == 5) ? y5 : (m == 6) ? y6 : y7;

    const int lane  = threadIdx.x;
    const int row   = rb * 16 + (lane & 15);
    const int khalf = (lane >> 4) * 2;          // 0 for lanes 0-15, 2 for 16-31

    v8f c = {};
    v2f b;  b.x = 1.0f;  b.y = 1.0f;

    const float* rowp = Y + (size_t)row * E_ + khalf;
    #pragma unroll 4
    for (int k0 = 0; k0 < E_; k0 += 4) {
        v2f a = *(const v2f*)(rowp + k0);       // K = k0+khalf, k0+khalf+1
        c = __builtin_amdgcn_wmma_f32_16x16x4_f32(
                /*neg_a=*/false, a, /*neg_b=*/false, b,
                /*c_mod=*/(short)0, c, /*reuse_a=*/false, /*reuse_b=*/false);
    }

    float* o = ws + (size_t)m * E_ + rb * 16;
    if (lane == 0) {
        #pragma unroll
        for (int v = 0; v < 8; ++v) o[v] = c[v];        // M = 0..7 (N=0 column)
    } else if (lane == 16) {
        #pragma unroll
        for (int v = 0; v < 8; ++v) o[8 + v] = c[v];    // M = 8..15 (N=0 column)
    }
}

// ---------------------------------------------------------------------------
// Kernel 2: memory-bound streaming pass. 1 thread = 4 consecutive e values.
// ---------------------------------------------------------------------------
__device__ __forceinline__ v4f sshrink4(v4f v) {
    v4f r;
    #pragma unroll
    for (int i = 0; i < 4; ++i) {
        float x = v[i];
        r[i] = (x > LAMBDA_) ? (x - LAMBDA_) : ((x < -LAMBDA_) ? (x + LAMBDA_) : 0.0f);
    }
    return r;
}

__device__ __forceinline__ void cm(v4f xr, v4f xi, v4f sr, v4f si,
                                   v4f& re, v4f& im) {
    re = sshrink4(xr * sr - xi * si);
    im = sshrink4(xr * si + xi * sr);
}

__global__ void __launch_bounds__(256) octmlp_main(
    const float* __restrict__ xr, const float* __restrict__ xi,
    const float* __restrict__ ws,
    const float* __restrict__ bR1, const float* __restrict__ bI1,
    const float* __restrict__ bR2, const float* __restrict__ bI2,
    const float* __restrict__ bR3, const float* __restrict__ bI3,
    const float* __restrict__ bR4, const float* __restrict__ bI4,
    float* __restrict__ out)
{
    const int QE  = E_ / 4;  // 128 float4 quads per row
    int gid = blockIdx.x * blockDim.x + threadIdx.x;
    int bs  = gid / QE;
    int e4  = (gid - bs * QE) * 4;
    if (bs >= BS_) return;

    const float* xrb = xr + (size_t)bs * 4 * E_ + e4;
    const float* xib = xi + (size_t)bs * 4 * E_ + e4;

    // Streaming reads: read-once, non-temporal.
    v4f x1r = __builtin_nontemporal_load((const v4f*)(xrb + 0 * E_));
    v4f x2r = __builtin_nontemporal_load((const v4f*)(xrb + 1 * E_));
    v4f x3r = __builtin_nontemporal_load((const v4f*)(xrb + 2 * E_));
    v4f x4r = __builtin_nontemporal_load((const v4f*)(xrb + 3 * E_));
    v4f x1i = __builtin_nontemporal_load((const v4f*)(xib + 0 * E_));
    v4f x2i = __builtin_nontemporal_load((const v4f*)(xib + 1 * E_));
    v4f x3i = __builtin_nontemporal_load((const v4f*)(xib + 2 * E_));
    v4f x4i = __builtin_nontemporal_load((const v4f*)(xib + 3 * E_));

    // Broadcast vectors (32 KB working set): regular temporal loads.
    v4f r1 = *(const v4f*)(ws + 0 * E_ + e4);
    v4f i1 = *(const v4f*)(ws + 1 * E_ + e4);
    v4f r2 = *(const v4f*)(ws + 2 * E_ + e4);
    v4f i2 = *(const v4f*)(ws + 3 * E_ + e4);
    v4f r3 = *(const v4f*)(ws + 4 * E_ + e4);
    v4f i3 = *(const v4f*)(ws + 5 * E_ + e4);
    v4f r4 = *(const v4f*)(ws + 6 * E_ + e4);
    v4f i4 = *(const v4f*)(ws + 7 * E_ + e4);
    v4f ni2 = -i2, ni3 = -i3, ni4 = -i4;

    v4f tre, tim;
    v4f a1re, a1im, a2re, a2im, a3re, a3im, a4re, a4im;

    // a1 = cm(x1,r1,i1) - cm(x2,r2,-i2) - cm(x3,r3,-i3) - cm(x4,r4,-i4)
    cm(x1r, x1i, r1, i1, a1re, a1im);
    cm(x2r, x2i, r2, ni2, tre, tim);  a1re -= tre;  a1im -= tim;
    cm(x3r, x3i, r3, ni3, tre, tim);  a1re -= tre;  a1im -= tim;
    cm(x4r, x4i, r4, ni4, tre, tim);  a1re -= tre;  a1im -= tim;

    // a2 = cm(x2,r1,i1) + cm(x1,r2,i2) + cm(x4,r3,-i3) - cm(x3,r4,-i4)
    cm(x2r, x2i, r1, i1, a2re, a2im);
    cm(x1r, x1i, r2, i2, tre, tim);   a2re += tre;  a2im += tim;
    cm(x4r, x4i, r3, ni3, tre, tim);  a2re += tre;  a2im += tim;
    cm(x3r, x3i, r4, ni4, tre, tim);  a2re -= tre;  a2im -= tim;

    // a3 = cm(x3,r1,i1) - cm(x4,r2,-i2) + cm(x1,r3,i3) + cm(x2,r4,-i4)
    cm(x3r, x3i, r1, i1, a3re, a3im);
    cm(x4r, x4i, r2, ni2, tre, tim);  a3re -= tre;  a3im -= tim;
    cm(x1r, x1i, r3, i3, tre, tim);   a3re += tre;  a3im += tim;
    cm(x2r, x2i, r4, ni4, tre, tim);  a3re += tre;  a3im += tim;

    // a4 = cm(x4,r1,i1) + cm(x3,r2,-i2) - cm(x2,r3,-i3) + cm(x1,r4,i4)
    cm(x4r, x4i, r1, i1, a4re, a4im);
    cm(x3r, x3i, r2, ni2, tre, tim);  a4re += tre;  a4im += tim;
    cm(x2r, x2i, r3, ni3, tre, tim);  a4re -= tre;  a4im -= tim;
    cm(x1r, x1i, r4, i4, tre, tim);   a4re += tre;  a4im += tim;

    v4f vbR1 = *(const v4f*)(bR1 + e4), vbI1 = *(const v4f*)(bI1 + e4);
    v4f vbR2 = *(const v4f*)(bR2 + e4), vbI2 = *(const v4f*)(bI2 + e4);
    v4f vbR3 = *(const v4f*)(bR3 + e4), vbI3 = *(const v4f*)(bI3 + e4);
    v4f vbR4 = *(const v4f*)(bR4 + e4), vbI4 = *(const v4f*)(bI4 + e4);

    // Output layout: out[((k*2 + c)*BS + bs)*E + e]. Write-once -> NT stores.
    const size_t plane = (size_t)BS_ * E_;
    float* o = out + (size_t)bs * E_ + e4;
    __builtin_nontemporal_store(sshrink4(a1re + vbR1), (v4f*)(o + 0 * plane));
    __builtin_nontemporal_store(sshrink4(a1im + vbI1), (v4f*)(o + 1 * plane));
    __builtin_nontemporal_store(sshrink4(a2re + vbR2), (v4f*)(o + 2 * plane));
    __builtin_nontemporal_store(sshrink4(a2im + vbI2), (v4f*)(o + 3 * plane));
    __builtin_nontemporal_store(sshrink4(a3re + vbR3), (v4f*)(o + 4 * plane));
    __builtin_nontemporal_store(sshrink4(a3im + vbI3), (v4f*)(o + 5 * plane));
    __builtin_nontemporal_store(sshrink4(a4re + vbR4), (v4f*)(o + 6 * plane));
    __builtin_nontemporal_store(sshrink4(a4im + vbI4), (v4f*)(o + 7 * plane));
}

extern "C" void kernel_launch(void* const* d_in, const int* in_sizes, int n_in,
                              void* d_out, int out_size, void* d_ws, size_t ws_size,
                              hipStream_t stream) {
    const float* xr = (const float*)d_in[0];
    const float* xi = (const float*)d_in[1];
    float* ws = (float*)d_ws;   // 8 * 512 f32 row sums: r1,i1,r2,i2,r3,i3,r4,i4

    // Pass 1: WMMA row-sum reduction of the 8 weight matrices.
    rowsum_wmma<<<8 * 32, 32, 0, stream>>>(
        (const float*)d_in[2], (const float*)d_in[3],
        (const float*)d_in[4], (const float*)d_in[5],
        (const float*)d_in[6], (const float*)d_in[7],
        (const float*)d_in[8], (const float*)d_in[9], ws);

    // Pass 2: streaming elementwise pass (memory-bound, ~256 MB traffic).
    const int total  = BS_ * (E_ / 4);           // 1,048,576 threads
    const int block  = 256;
    const int grid   = (total + block - 1) / block;
    octmlp_main<<<grid, block, 0, stream>>>(
        xr, xi, ws,
        (const float*)d_in[10], (const float*)d_in[11],
        (const float*)d_in[12], (const float*)d_in[13],
        (const float*)d_in[14], (const float*)d_in[15],
        (const float*)d_in[16], (const float*)d_in[17],
        (float*)d_out);
}